// Adder8Bit_43860206027280
// MI455X (gfx1250) — compile-verified
//
#include <hip/hip_runtime.h>

// ---------------------------------------------------------------------------
// 8-bit spike adder: out_bits = bits of (A + B + Cin), carry = bit 8.
// Pure HBM-streaming problem (~436 MB total, ~18.7 us roofline at 23.3 TB/s).
// Main kernel streams A/B through LDS with CDNA5 async global->LDS b128 DMA
// (double-buffered, ASYNCcnt-synchronized), non-temporal hints everywhere
// (one-shot 436 MB stream > 192 MB L2, zero reuse).
// ---------------------------------------------------------------------------

typedef unsigned int u32;
typedef unsigned long long u64;
typedef u32   v4u __attribute__((ext_vector_type(4)));
typedef float v4f __attribute__((ext_vector_type(4)));

#define TPB 256
#define TILES_PER_BLOCK 8

// f is the IEEE-754 bit pattern of 0.0f or 1.0f -> extract the bit.
__device__ __forceinline__ u32 bit_of(u32 f) { return (f >> 23) & 1u; }

// lo = row elements j=0..3 (bits 7..4, MSB first), hi = j=4..7 (bits 3..0)
__device__ __forceinline__ u32 pack_row(v4u lo, v4u hi) {
    u32 v;
    v  = bit_of(lo.x) << 7;
    v |= bit_of(lo.y) << 6;
    v |= bit_of(lo.z) << 5;
    v |= bit_of(lo.w) << 4;
    v |= bit_of(hi.x) << 3;
    v |= bit_of(hi.y) << 2;
    v |= bit_of(hi.z) << 1;
    v |= bit_of(hi.w);
    return v;
}

__device__ __forceinline__ void unpack_store(u32 s, v4f* __restrict__ outS,
                                             float* __restrict__ outC, size_t row) {
    v4f lo, hi;
    lo.x = (float)((s >> 7) & 1u);
    lo.y = (float)((s >> 6) & 1u);
    lo.z = (float)((s >> 5) & 1u);
    lo.w = (float)((s >> 4) & 1u);
    hi.x = (float)((s >> 3) & 1u);
    hi.y = (float)((s >> 2) & 1u);
    hi.z = (float)((s >> 1) & 1u);
    hi.w = (float)( s       & 1u);
    __builtin_nontemporal_store(lo, outS + 2 * row);
    __builtin_nontemporal_store(hi, outS + 2 * row + 1);
    __builtin_nontemporal_store((float)((s >> 8) & 1u), outC + row);
}

// CDNA5 async global->LDS 128-bit load (fire-and-forget, tracked by ASYNCcnt).
// th:TH_LOAD_NT: read-once stream, do not rinse through L2 on the way to LDS.
__device__ __forceinline__ void async_load_b128(u32 lds_off, u64 gaddr) {
    asm volatile("global_load_async_to_lds_b128 %0, %1, off th:TH_LOAD_NT"
                 :: "v"(lds_off), "v"(gaddr) : "memory");
}

// ---------------------------------------------------------------------------
// Main streaming kernel: 8 tiles of 256 rows per block, double-buffered LDS.
// Each thread stages and consumes only its own 64 B slot -> no barriers;
// per-wave in-order ASYNCcnt completion is the only sync needed.
// ---------------------------------------------------------------------------
__global__ __launch_bounds__(TPB) void adder8_async(
        const v4u* __restrict__ A, const v4u* __restrict__ B,
        const u32* __restrict__ Cin,
        v4f* __restrict__ OutS, float* __restrict__ OutC) {
    __shared__ v4u lds[2][TPB][4];   // 2 x 16 KB double buffer (32 KB/block)
    const int tid      = threadIdx.x;
    const int tileBase = blockIdx.x * TILES_PER_BLOCK;

    auto stage = [&](int buf, int tile) {
        size_t row = (size_t)(tileBase + tile) * TPB + tid;
        u32 l = (u32)(u64)(__attribute__((address_space(3))) char*)
                    ((char*)&lds[buf][tid][0]);
        u64 ga = (u64)(const char*)(A + 2 * row);
        u64 gb = (u64)(const char*)(B + 2 * row);
        async_load_b128(l,      ga);
        async_load_b128(l + 16, ga + 16);
        async_load_b128(l + 32, gb);
        async_load_b128(l + 48, gb + 16);
    };

    stage(0, 0);
#pragma unroll
    for (int t = 0; t < TILES_PER_BLOCK; ++t) {
        const int buf = t & 1;
        if (t + 1 < TILES_PER_BLOCK) {
            stage(buf ^ 1, t + 1);                             // prefetch next tile
            asm volatile("s_wait_asynccnt 0x4" ::: "memory");  // current tile done
        } else {
            asm volatile("s_wait_asynccnt 0x0" ::: "memory");
        }
        size_t row = (size_t)(tileBase + t) * TPB + tid;
        u32 cin = bit_of(__builtin_nontemporal_load(Cin + row));
        v4u a0 = lds[buf][tid][0];
        v4u a1 = lds[buf][tid][1];
        v4u b0 = lds[buf][tid][2];
        v4u b1 = lds[buf][tid][3];
        u32 s = pack_row(a0, a1) + pack_row(b0, b1) + cin;
        unpack_store(s, OutS, OutC, row);
    }
}

// ---------------------------------------------------------------------------
// Remainder kernel (direct NT loads) for rows not covered by full tiles.
// ---------------------------------------------------------------------------
__global__ __launch_bounds__(TPB) void adder8_direct(
        const v4u* __restrict__ A, const v4u* __restrict__ B,
        const u32* __restrict__ Cin,
        v4f* __restrict__ OutS, float* __restrict__ OutC,
        int nrows, int row0) {
    int i = row0 + blockIdx.x * blockDim.x + threadIdx.x;
    if (i >= nrows) return;
    size_t r = (size_t)i;
    v4u a0 = __builtin_nontemporal_load(A + 2 * r);
    v4u a1 = __builtin_nontemporal_load(A + 2 * r + 1);
    v4u b0 = __builtin_nontemporal_load(B + 2 * r);
    v4u b1 = __builtin_nontemporal_load(B + 2 * r + 1);
    u32 cin = bit_of(__builtin_nontemporal_load(Cin + r));
    u32 s = pack_row(a0, a1) + pack_row(b0, b1) + cin;
    unpack_store(s, OutS, OutC, r);
}

extern "C" void kernel_launch(void* const* d_in, const int* in_sizes, int n_in,
                              void* d_out, int out_size, void* d_ws, size_t ws_size,
                              hipStream_t stream) {
    const v4u* A   = (const v4u*)d_in[0];   // [N,8] f32 in {0,1}, MSB first
    const v4u* B   = (const v4u*)d_in[1];   // [N,8] f32
    const u32* Cin = (const u32*)d_in[2];   // [N,1] f32 (bit patterns)
    const int nrows = in_sizes[0] / 8;

    float* out  = (float*)d_out;
    v4f*   OutS = (v4f*)out;                // N*8 sum bits (2 x v4f / row)
    float* OutC = out + (size_t)nrows * 8;  // N carry bits

    const int rows_per_block = TPB * TILES_PER_BLOCK;
    const int nblocks = nrows / rows_per_block;
    const int covered = nblocks * rows_per_block;

    if (nblocks > 0) {
        adder8_async<<<nblocks, TPB, 0, stream>>>(A, B, Cin, OutS, OutC);
    }
    if (covered < nrows) {
        const int rem = nrows - covered;
        adder8_direct<<<(rem + TPB - 1) / TPB, TPB, 0, stream>>>(
            A, B, Cin, OutS, OutC, nrows, covered);
    }
}